// SS3Dv0_30167850287874
// MI455X (gfx1250) — compile-verified
//
#include <hip/hip_runtime.h>

typedef float f32x2 __attribute__((ext_vector_type(2)));
typedef float f32x8 __attribute__((ext_vector_type(8)));

#define WMMA4(a, b, c) \
  __builtin_amdgcn_wmma_f32_16x16x4_f32(false, (a), false, (b), (short)0, (c), false, false)

__device__ __forceinline__ float siluf(float x)     { return x / (1.f + __expf(-x)); }
__device__ __forceinline__ float softplusf(float x) { return (x > 20.f) ? x : __logf(1.f + __expf(x)); }

// Sequence-position -> natural linear index for the 6 base scan orders.
// Natural index p = (h*16 + w)*8 + d  with (H,W,D)=(16,16,8).
__device__ __forceinline__ int pos_of(int k6, int t) {
  switch (k6) {
    case 0: return t;                                                          // (H,W,D)
    case 1: { int w = t & 15, d = (t >> 4) & 7,  h = t >> 7; return ((h << 4) + w) * 8 + d; } // (H,D,W)
    case 2: { int d = t & 7,  h = (t >> 3) & 15, w = t >> 7; return ((h << 4) + w) * 8 + d; } // (W,H,D)
    case 3: { int h = t & 15, d = (t >> 4) & 7,  w = t >> 7; return ((h << 4) + w) * 8 + d; } // (W,D,H)
    case 4: { int w = t & 15, h = (t >> 4) & 15, d = t >> 8; return ((h << 4) + w) * 8 + d; } // (D,H,W)
    default:{ int h = t & 15, w = (t >> 4) & 15, d = t >> 8; return ((h << 4) + w) * 8 + d; } // (D,W,H)
  }
}

// ---------------------------------------------------------------------------
// 1) in_proj: C[4096,384] = X[4096,96] * W^T, split into xin (b,c,l) and silu(z) (b,l,c)
// ---------------------------------------------------------------------------
__global__ void ss3d_inproj(const float* __restrict__ X, const float* __restrict__ W,
                            float* __restrict__ xin, float* __restrict__ z) {
  int gw = (blockIdx.x * blockDim.x + threadIdx.x) >> 5;
  int lane = threadIdx.x & 31, hw = lane >> 4, lm = lane & 15;
  int mt = gw / 24, nt = gw % 24;
  int m = mt * 16 + lm, n = nt * 16 + lm;
  f32x8 acc;
#pragma unroll
  for (int i = 0; i < 8; i++) acc[i] = 0.f;
  for (int k0 = 0; k0 < 96; k0 += 4) {
    int ka = k0 + 2 * hw;
    f32x2 a, b;
    a.x = X[m * 96 + ka]; a.y = X[m * 96 + ka + 1];
    b.x = W[n * 96 + ka]; b.y = W[n * 96 + ka + 1];
    acc = WMMA4(a, b, acc);
  }
#pragma unroll
  for (int i = 0; i < 8; i++) {
    int mm = mt * 16 + 8 * hw + i;
    int bb = mm >> 11, l = mm & 2047;
    float v = acc[i];
    if (n < 192) xin[(bb * 192 + n) * 2048 + l] = v;
    else         z[mm * 192 + (n - 192)] = siluf(v);
  }
}

// ---------------------------------------------------------------------------
// 2) depthwise conv3d 3x3x3 (SAME) + bias + silu -> xc (b,c,l)
// ---------------------------------------------------------------------------
__global__ void ss3d_conv(const float* __restrict__ xin, const float* __restrict__ cw,
                          const float* __restrict__ cb, float* __restrict__ xc) {
  int idx = blockIdx.x * 256 + threadIdx.x;
  if (idx >= 2 * 192 * 2048) return;
  int l = idx & 2047;
  int ch = (idx >> 11) % 192;
  int b = idx / (192 * 2048);
  int h = l >> 7, w = (l >> 3) & 15, d = l & 7;
  const float* base = xin + (b * 192 + ch) * 2048;
  float s = cb[ch];
#pragma unroll
  for (int i = 0; i < 3; i++) {
    int hh = h + i - 1; if ((unsigned)hh >= 16u) continue;
#pragma unroll
    for (int j = 0; j < 3; j++) {
      int ww = w + j - 1; if ((unsigned)ww >= 16u) continue;
#pragma unroll
      for (int kk = 0; kk < 3; kk++) {
        int dd = d + kk - 1; if ((unsigned)dd >= 8u) continue;
        s += base[((hh << 4) + ww) * 8 + dd] * cw[ch * 27 + i * 9 + j * 3 + kk];
      }
    }
  }
  xc[idx] = siluf(s);
}

// ---------------------------------------------------------------------------
// 3) x_dbl per (b,k): [2048 x 38] = gathered-xs[2048 x 192] * Wk^T. Stored by
//    sequence position (reversal folded into the gather index).
// ---------------------------------------------------------------------------
__global__ void ss3d_xdbl(const float* __restrict__ xc, const float* __restrict__ xpw,
                          float* __restrict__ xdbl) {
  int gw = (blockIdx.x * blockDim.x + threadIdx.x) >> 5;
  int lane = threadIdx.x & 31, hw = lane >> 4, lm = lane & 15;
  int nt = gw % 3; int rem = gw / 3;
  int mt = rem % 128; rem /= 128;
  int k = rem % 12; int b = rem / 12;
  int t = mt * 16 + lm;
  int tp = (k < 6) ? t : (2047 - t);
  int p = pos_of(k % 6, tp);
  int j = nt * 16 + lm;
  bool jok = (j < 38);
  f32x8 acc;
#pragma unroll
  for (int i = 0; i < 8; i++) acc[i] = 0.f;
  for (int k0 = 0; k0 < 192; k0 += 4) {
    int ka = k0 + 2 * hw;
    f32x2 a, bf;
    a.x = xc[(b * 192 + ka) * 2048 + p];
    a.y = xc[(b * 192 + ka + 1) * 2048 + p];
    bf.x = jok ? xpw[(k * 38 + j) * 192 + ka]     : 0.f;
    bf.y = jok ? xpw[(k * 38 + j) * 192 + ka + 1] : 0.f;
    acc = WMMA4(a, bf, acc);
  }
  if (jok) {
#pragma unroll
    for (int i = 0; i < 8; i++) {
      int tt = mt * 16 + 8 * hw + i;
      xdbl[((b * 12 + k) * 2048 + tt) * 38 + j] = acc[i];
    }
  }
}

// ---------------------------------------------------------------------------
// 4) down proj: xd[(b,c),j] = xc[(b,c),:] . down_w[j,:] + down_b[j]
// ---------------------------------------------------------------------------
__global__ void ss3d_down(const float* __restrict__ xc, const float* __restrict__ dw,
                          const float* __restrict__ db, float* __restrict__ xd) {
  int gw = (blockIdx.x * blockDim.x + threadIdx.x) >> 5;
  int lane = threadIdx.x & 31, hw = lane >> 4, lm = lane & 15;
  int mt = gw / 8, nt = gw % 8;
  int m = mt * 16 + lm, n = nt * 16 + lm;
  f32x8 acc;
#pragma unroll
  for (int i = 0; i < 8; i++) acc[i] = 0.f;
  for (int k0 = 0; k0 < 2048; k0 += 4) {
    int ka = k0 + 2 * hw;
    f32x2 a, b;
    a.x = xc[m * 2048 + ka]; a.y = xc[m * 2048 + ka + 1];
    b.x = dw[n * 2048 + ka]; b.y = dw[n * 2048 + ka + 1];
    acc = WMMA4(a, b, acc);
  }
#pragma unroll
  for (int i = 0; i < 8; i++) {
    int mm = mt * 16 + 8 * hw + i;
    xd[mm * 128 + n] = acc[i] + db[n];
  }
}

// ---------------------------------------------------------------------------
// 5) channel x_dbl: [384 x 40] = xd[384 x 128] * xpc^T
// ---------------------------------------------------------------------------
__global__ void ss3d_xdblc(const float* __restrict__ xd, const float* __restrict__ xpc,
                           float* __restrict__ xdblc) {
  int gw = (blockIdx.x * blockDim.x + threadIdx.x) >> 5;
  int lane = threadIdx.x & 31, hw = lane >> 4, lm = lane & 15;
  int mt = gw / 3, nt = gw % 3;
  int m = mt * 16 + lm, n = nt * 16 + lm;
  bool nok = (n < 40);
  f32x8 acc;
#pragma unroll
  for (int i = 0; i < 8; i++) acc[i] = 0.f;
  for (int k0 = 0; k0 < 128; k0 += 4) {
    int ka = k0 + 2 * hw;
    f32x2 a, b;
    a.x = xd[m * 128 + ka]; a.y = xd[m * 128 + ka + 1];
    b.x = nok ? xpc[n * 128 + ka]     : 0.f;
    b.y = nok ? xpc[n * 128 + ka + 1] : 0.f;
    acc = WMMA4(a, b, acc);
  }
  if (nok) {
#pragma unroll
    for (int i = 0; i < 8; i++) {
      int mm = mt * 16 + 8 * hw + i;
      xdblc[mm * 40 + n] = acc[i];
    }
  }
}

// ---------------------------------------------------------------------------
// 6) channel selective scan: 192 steps, 128 lanes/block, one block per batch
// ---------------------------------------------------------------------------
__global__ void ss3d_chscan(const float* __restrict__ xd, const float* __restrict__ xdblc,
                            const float* __restrict__ dtcw, const float* __restrict__ dtcb,
                            const float* __restrict__ Alc, const float* __restrict__ Dsc,
                            float* __restrict__ yc) {
  int b = blockIdx.x, j = threadIdx.x;
  float w[8];
#pragma unroll
  for (int r = 0; r < 8; r++) w[r] = dtcw[j * 8 + r];
  float bias = dtcb[j];
  float An[16], h[16];
#pragma unroll
  for (int n = 0; n < 16; n++) { An[n] = -__expf(Alc[j * 16 + n]); h[n] = 0.f; }
  float Dc = Dsc[j];
  for (int c = 0; c < 192; c++) {
    const float* row = xdblc + (b * 192 + c) * 40;
    float dtr = bias;
#pragma unroll
    for (int r = 0; r < 8; r++) dtr += w[r] * row[r];
    float dt = softplusf(dtr);
    float u = xd[(b * 192 + c) * 128 + j];
    float du = dt * u, y = 0.f;
#pragma unroll
    for (int n = 0; n < 16; n++) {
      float e = __expf(dt * An[n]);
      h[n] = h[n] * e + du * row[8 + n];
      y += h[n] * row[24 + n];
    }
    yc[(b * 192 + c) * 128 + j] = y + Dc * u;
  }
}

// ---------------------------------------------------------------------------
// 7) up proj (initializes ysum): ysum[(b,p),c] = up_w[p,:] . yc[(b,c),:] + up_b[p]
// ---------------------------------------------------------------------------
__global__ void ss3d_up(const float* __restrict__ uw, const float* __restrict__ ub,
                        const float* __restrict__ yc, float* __restrict__ ysum) {
  int gw = (blockIdx.x * blockDim.x + threadIdx.x) >> 5;
  int lane = threadIdx.x & 31, hw = lane >> 4, lm = lane & 15;
  int b = gw / (128 * 12); int rem = gw % (128 * 12);
  int mt = rem / 12, nt = rem % 12;
  int m = mt * 16 + lm, n = nt * 16 + lm;
  f32x8 acc;
#pragma unroll
  for (int i = 0; i < 8; i++) acc[i] = 0.f;
  for (int k0 = 0; k0 < 128; k0 += 4) {
    int ka = k0 + 2 * hw;
    f32x2 a, bf;
    a.x = uw[m * 128 + ka]; a.y = uw[m * 128 + ka + 1];
    bf.x = yc[(b * 192 + n) * 128 + ka];
    bf.y = yc[(b * 192 + n) * 128 + ka + 1];
    acc = WMMA4(a, bf, acc);
  }
#pragma unroll
  for (int i = 0; i < 8; i++) {
    int mm = mt * 16 + 8 * hw + i;
    ysum[(b * 2048 + mm) * 192 + n] = acc[i] + ub[mm];
  }
}

// ---------------------------------------------------------------------------
// 8) spatial selective scan: block per (b,k), lane per channel c; atomically
//    accumulates all 12 directions into ysum (b,L,C)
// ---------------------------------------------------------------------------
__global__ void ss3d_sscan(const float* __restrict__ xc, const float* __restrict__ xdbl,
                           const float* __restrict__ dtw_, const float* __restrict__ dtb,
                           const float* __restrict__ Alg, const float* __restrict__ Dsg,
                           float* __restrict__ ysum) {
  int bk = blockIdx.x;
  int b = bk / 12, k = bk % 12;
  int c = threadIdx.x;
  float w[6];
#pragma unroll
  for (int r = 0; r < 6; r++) w[r] = dtw_[(k * 192 + c) * 6 + r];
  float bias = dtb[k * 192 + c];
  float An[16], h[16];
#pragma unroll
  for (int n = 0; n < 16; n++) { An[n] = -__expf(Alg[(k * 192 + c) * 16 + n]); h[n] = 0.f; }
  float Dv = Dsg[k * 192 + c];
  const float* xcb = xc + (b * 192 + c) * 2048;
  const float* xrow = xdbl + (size_t)(b * 12 + k) * 2048 * 38;
  int k6 = k % 6;
  bool rev = (k >= 6);
  for (int s = 0; s < 2048; s++) {
    const float* row = xrow + s * 38;
    float dtr = bias;
#pragma unroll
    for (int r = 0; r < 6; r++) dtr += w[r] * row[r];
    float dt = softplusf(dtr);
    int tp = rev ? (2047 - s) : s;
    int p = pos_of(k6, tp);
    float u = xcb[p];
    float du = dt * u, y = 0.f;
#pragma unroll
    for (int n = 0; n < 16; n++) {
      float e = __expf(dt * An[n]);
      h[n] = h[n] * e + du * row[6 + n];
      y += h[n] * row[22 + n];
    }
    atomicAdd(&ysum[(b * 2048 + p) * 192 + c], y + Dv * u);
  }
}

// ---------------------------------------------------------------------------
// 9) layernorm over C=192 + gate by silu(z), in place in ysum
// ---------------------------------------------------------------------------
__global__ void ss3d_norm(float* __restrict__ ysum, const float* __restrict__ z,
                          const float* __restrict__ nw, const float* __restrict__ nb) {
  __shared__ float s1[192], s2[192];
  int row = blockIdx.x, c = threadIdx.x;
  float v = ysum[row * 192 + c];
  s1[c] = v; s2[c] = v * v;
  __syncthreads();
  for (int off = 96; off >= 3; off >>= 1) {
    if (c < off) { s1[c] += s1[c + off]; s2[c] += s2[c + off]; }
    __syncthreads();
  }
  if (c == 0) {
    s1[0] = (s1[0] + s1[1] + s1[2]) * (1.f / 192.f);
    s2[0] = (s2[0] + s2[1] + s2[2]) * (1.f / 192.f);
  }
  __syncthreads();
  float mu = s1[0];
  float var = s2[0] - mu * mu;
  float yn = (v - mu) * rsqrtf(var + 1e-5f) * nw[c] + nb[c];
  ysum[row * 192 + c] = yn * z[row * 192 + c];
}

// ---------------------------------------------------------------------------
// 10) out proj: out[4096,96] = ysum[4096,192] * out_w^T
// ---------------------------------------------------------------------------
__global__ void ss3d_out(const float* __restrict__ ys, const float* __restrict__ ow,
                         float* __restrict__ out) {
  int gw = (blockIdx.x * blockDim.x + threadIdx.x) >> 5;
  int lane = threadIdx.x & 31, hw = lane >> 4, lm = lane & 15;
  int mt = gw / 6, nt = gw % 6;
  int m = mt * 16 + lm, n = nt * 16 + lm;
  f32x8 acc;
#pragma unroll
  for (int i = 0; i < 8; i++) acc[i] = 0.f;
  for (int k0 = 0; k0 < 192; k0 += 4) {
    int ka = k0 + 2 * hw;
    f32x2 a, b;
    a.x = ys[m * 192 + ka]; a.y = ys[m * 192 + ka + 1];
    b.x = ow[n * 192 + ka]; b.y = ow[n * 192 + ka + 1];
    acc = WMMA4(a, b, acc);
  }
#pragma unroll
  for (int i = 0; i < 8; i++) {
    int mm = mt * 16 + 8 * hw + i;
    out[mm * 96 + n] = acc[i];
  }
}

extern "C" void kernel_launch(void* const* d_in, const int* in_sizes, int n_in,
                              void* d_out, int out_size, void* d_ws, size_t ws_size,
                              hipStream_t stream) {
  const float* x    = (const float*)d_in[0];
  const float* ipw  = (const float*)d_in[1];
  const float* cw   = (const float*)d_in[2];
  const float* cb   = (const float*)d_in[3];
  const float* xpw  = (const float*)d_in[4];
  const float* dtw  = (const float*)d_in[5];
  const float* dtb  = (const float*)d_in[6];
  const float* alg  = (const float*)d_in[7];
  const float* ds   = (const float*)d_in[8];
  const float* xpc  = (const float*)d_in[9];
  const float* dtcw = (const float*)d_in[10];
  const float* dtcb = (const float*)d_in[11];
  const float* alc  = (const float*)d_in[12];
  const float* dsc  = (const float*)d_in[13];
  const float* dw   = (const float*)d_in[14];
  const float* db   = (const float*)d_in[15];
  const float* uw   = (const float*)d_in[16];
  const float* ub   = (const float*)d_in[17];
  const float* nw   = (const float*)d_in[18];
  const float* nb   = (const float*)d_in[19];
  const float* ow   = (const float*)d_in[20];
  float* out = (float*)d_out;

  float* ws    = (float*)d_ws;
  float* xin   = ws;                 // (b,192,2048)     786432
  float* z     = xin   + 786432;     // (b*2048,192)     786432
  float* xc    = z     + 786432;     // (b,192,2048)     786432
  float* xdbl  = xc    + 786432;     // (b,12,2048,38)   1867776
  float* xd    = xdbl  + 1867776;    // (b,192,128)      49152
  float* xdblc = xd    + 49152;      // (b,192,40)       15360
  float* yc    = xdblc + 15360;      // (b,192,128)      49152
  float* ysum  = yc    + 49152;      // (b,2048,192)     786432

  ss3d_inproj<<<768,  256, 0, stream>>>(x, ipw, xin, z);
  ss3d_conv  <<<3072, 256, 0, stream>>>(xin, cw, cb, xc);
  ss3d_xdbl  <<<1152, 256, 0, stream>>>(xc, xpw, xdbl);
  ss3d_down  <<<24,   256, 0, stream>>>(xc, dw, db, xd);
  ss3d_xdblc <<<9,    256, 0, stream>>>(xd, xpc, xdblc);
  ss3d_chscan<<<2,    128, 0, stream>>>(xd, xdblc, dtcw, dtcb, alc, dsc, yc);
  ss3d_up    <<<384,  256, 0, stream>>>(uw, ub, yc, ysum);
  ss3d_sscan <<<24,   192, 0, stream>>>(xc, xdbl, dtw, dtb, alg, ds, ysum);
  ss3d_norm  <<<4096, 192, 0, stream>>>(ysum, z, nw, nb);
  ss3d_out   <<<192,  256, 0, stream>>>(ysum, ow, out);
}